// Kernel_59347858096260
// MI455X (gfx1250) — compile-verified
//
#include <hip/hip_runtime.h>

// ---- problem constants (from reference) ----
#define IC_   32
#define OC_   64
#define DIM_  64
#define OD_   62
constexpr int CSTR = DIM_ * DIM_ * DIM_;   // 262144 floats between input channels
constexpr int ZSTR = DIM_ * DIM_;          // 4096
constexpr int YSTR = DIM_;                 // 64
constexpr int OSP  = OD_ * OD_ * OD_;      // 238328 floats per output channel
constexpr int NPACK = 27 * 8 * 4 * 32 * 2; // 55296 packed weight floats (221 KB)

typedef float v2f __attribute__((ext_vector_type(2)));
typedef float v8f __attribute__((ext_vector_type(8)));
typedef int   v4i __attribute__((ext_vector_type(4)));
typedef __attribute__((address_space(1))) v4i gv4i;   // global
typedef __attribute__((address_space(3))) v4i lv4i;   // LDS

#if defined(__has_builtin)
#  if __has_builtin(__builtin_amdgcn_global_load_async_to_lds_b128)
#    define HAVE_ASYNC_LDS 1
#  endif
#endif
#ifndef HAVE_ASYNC_LDS
#  define HAVE_ASYNC_LDS 0
#endif

// ---------------------------------------------------------------------------
// Kernel 1: repack W (OIDHW) into WMMA A-fragment order:
//   Apack[tap][icc][mt][lane][j],  tap=(kz*3+ky)*3+kx, icc=ic/4, mt=oc/16,
//   lane = ((ic>>1)&1)*16 + (oc & 15),  j = ic & 1
// so a wave fetches one 16x4 f32 A fragment with a single ds_load_b64/lane.
// ---------------------------------------------------------------------------
__global__ void pack_weights(const float* __restrict__ Wsrc, float* __restrict__ Apack) {
    int idx = blockIdx.x * 256 + threadIdx.x;
    if (idx >= NPACK) return;
    int j    =  idx        & 1;
    int lane = (idx >> 1)  & 31;
    int mt   = (idx >> 6)  & 3;
    int icc  = (idx >> 8)  & 7;
    int tap  =  idx >> 11;           // 0..26
    int row  = lane & 15;
    int ps   = lane >> 4;
    int oc   = mt * 16 + row;
    int ic   = icc * 4 + ps * 2 + j;
    // W layout OIDHW: ((oc*IC + ic)*27 + tap)
    Apack[idx] = Wsrc[(oc * IC_ + ic) * 27 + tap];
}

// ---------------------------------------------------------------------------
// Stage one tap's 8 KB of A-fragments into an LDS buffer.
// Async path: GLOBAL_LOAD_ASYNC_TO_LDS_B128, 128 thr x 4 x 16B, ASYNCcnt.
// ---------------------------------------------------------------------------
__device__ __forceinline__ void stage_tap(const float* __restrict__ Apack,
                                          const float* __restrict__ Wraw,
                                          float4* __restrict__ dstBuf,
                                          int tap, int t, int packed) {
    if (packed) {
#if HAVE_ASYNC_LDS
        const float4* src = (const float4*)(Apack + tap * 2048);
        #pragma unroll
        for (int i = 0; i < 4; ++i) {
            __builtin_amdgcn_global_load_async_to_lds_b128(
                (gv4i*)(src + t + 128 * i),
                (lv4i*)(dstBuf + t + 128 * i),
                0, 0);
        }
#else
        const float4* src = (const float4*)(Apack + tap * 2048);
        #pragma unroll
        for (int i = 0; i < 4; ++i) dstBuf[t + 128 * i] = src[t + 128 * i];
#endif
    } else {
        // slow fallback: gather straight from OIDHW weights
        float* lds = (float*)dstBuf;
        #pragma unroll
        for (int i = 0; i < 16; ++i) {
            int e   = t + 128 * i;         // e = oc*32 + ic
            int oc  = e >> 5, ic = e & 31;
            int icc = ic >> 2, j = ic & 1, p2 = (ic >> 1) & 1;
            int mt  = oc >> 4, row = oc & 15;
            int l2  = p2 * 16 + row;
            lds[((icc * 4 + mt) * 32 + l2) * 2 + j] = Wraw[e * 27 + tap];
        }
    }
}

__device__ __forceinline__ void wait_stage(int packed) {
#if HAVE_ASYNC_LDS
    if (packed) {
#  if __has_builtin(__builtin_amdgcn_s_wait_asynccnt)
        __builtin_amdgcn_s_wait_asynccnt(0);
#  else
        asm volatile("s_wait_asynccnt 0x0" ::: "memory");
#  endif
    }
#else
    (void)packed;
#endif
}

// ---------------------------------------------------------------------------
// Kernel 2: implicit-GEMM conv3d with V_WMMA_F32_16X16X4_F32.
// Block = 128 threads (4 waves). Block owns one (b, od, oh) output row;
// wave w computes OC(64) x ow[16w .. 16w+15].  K loop: 27 taps x 8 ic-chunks,
// A staged per-tap into a double-buffered 2x8KB LDS region (async, overlapped
// with the current tap's 32 WMMAs), one barrier per tap.
// ---------------------------------------------------------------------------
__global__ __launch_bounds__(128) void conv3d_wmma(
    const float* __restrict__ x,
    const float* __restrict__ Wraw,
    const float* __restrict__ Apack,
    float* __restrict__ out,
    int packed)
{
    __shared__ float4 ldsA4[2][512];              // 2 x 8 KB double buffer

    const int t    = threadIdx.x;
    const int lane = t & 31;
    const int wid  = t >> 5;        // wave 0..3
    const int n    = lane & 15;     // N column within 16-wide tile
    const int ps   = lane >> 4;     // K-pair selector (0: K=0,1  1: K=2,3)

    int bidx = blockIdx.x;          // 0 .. 2*62*62-1
    const int oh  = bidx % OD_;  bidx /= OD_;
    const int od  = bidx % OD_;
    const int b   = bidx / OD_;

    const int ow0 = wid * 16;
    const float* xb = x + (size_t)b * IC_ * CSTR;

    v8f acc[4];
    #pragma unroll
    for (int mt = 0; mt < 4; ++mt) acc[mt] = (v8f){};

    stage_tap(Apack, Wraw, ldsA4[0], /*tap=*/0, t, packed);   // prologue

    for (int tap = 0; tap < 27; ++tap) {
        const int cur = tap & 1;
        const int kz = tap / 9;
        const int ky = (tap / 3) % 3;
        const int kx = tap % 3;

        wait_stage(packed);         // my wave's async writes for buf `cur` landed
        __syncthreads();            // everyone's writes visible; prev readers done

        if (tap + 1 < 27)           // overlap next tap's stage with this compute
            stage_tap(Apack, Wraw, ldsA4[cur ^ 1], tap + 1, t, packed);

        // B-fragment base address for this tap (clamped so padded lanes stay in-bounds)
        int w = ow0 + n + kx; if (w > DIM_ - 1) w = DIM_ - 1;
        const float* lb = xb + (size_t)(ps * 2) * CSTR
                             + (od + kz) * ZSTR + (oh + ky) * YSTR + w;

        const v2f* afrag = ((const v2f*)ldsA4[cur]) + lane;

        #pragma unroll
        for (int icc = 0; icc < 8; ++icc) {
            v2f bf;
            bf.x = lb[icc * 4 * CSTR];          // ic = icc*4 + ps*2
            bf.y = lb[icc * 4 * CSTR + CSTR];   // ic = icc*4 + ps*2 + 1
            const v2f* ap = afrag + icc * 128;
            // D = A(16x4) * B(4x16) + C, full fp32
            acc[0] = __builtin_amdgcn_wmma_f32_16x16x4_f32(
                         false, ap[0],   false, bf, (short)0, acc[0], false, false);
            acc[1] = __builtin_amdgcn_wmma_f32_16x16x4_f32(
                         false, ap[32],  false, bf, (short)0, acc[1], false, false);
            acc[2] = __builtin_amdgcn_wmma_f32_16x16x4_f32(
                         false, ap[64],  false, bf, (short)0, acc[2], false, false);
            acc[3] = __builtin_amdgcn_wmma_f32_16x16x4_f32(
                         false, ap[96],  false, bf, (short)0, acc[3], false, false);
        }
    }

    // Store: D layout = lane {0-15:M=v, 16-31:M=v+8} x VGPR v (ISA 7.12.2).
    // Non-temporal: output (122 MB) is write-once; keep x + weights hot in L2.
    const int ow = ow0 + n;
    if (ow < OD_) {
        const size_t obase = (size_t)b * OC_ * OSP + (size_t)od * (OD_ * OD_)
                           + (size_t)oh * OD_ + ow;
        #pragma unroll
        for (int mt = 0; mt < 4; ++mt) {
            #pragma unroll
            for (int v = 0; v < 8; ++v) {
                int oc = mt * 16 + v + 8 * ps;
                __builtin_nontemporal_store(acc[mt][v], &out[obase + (size_t)oc * OSP]);
            }
        }
    }
}

// ---------------------------------------------------------------------------
extern "C" void kernel_launch(void* const* d_in, const int* in_sizes, int n_in,
                              void* d_out, int out_size, void* d_ws, size_t ws_size,
                              hipStream_t stream) {
    const float* x  = (const float*)d_in[0];   // (2,32,64,64,64) fp32
    const float* Wv = (const float*)d_in[1];   // (64,32,3,3,3)  fp32
    float* out   = (float*)d_out;              // (2,64,62,62,62) fp32
    float* Apack = (float*)d_ws;

    const int packed = (ws_size >= (size_t)NPACK * sizeof(float)) ? 1 : 0;
    if (packed) {
        pack_weights<<<(NPACK + 255) / 256, 256, 0, stream>>>(Wv, Apack);
    }
    const int nblocks = 2 * OD_ * OD_;         // one block per (b, od, oh)
    conv3d_wmma<<<nblocks, 128, 0, stream>>>(x, Wv, Apack, out, packed);
}